// GATLayer_85280870629874
// MI455X (gfx1250) — compile-verified
//
#include <hip/hip_runtime.h>
#include <math.h>

typedef __attribute__((ext_vector_type(2))) float v2f;
typedef __attribute__((ext_vector_type(8))) float v8f;

#define BB   8
#define NN   2048
#define FF   128
#define LPITCH 130   // LDS pitch (floats): breaks 64-bank conflict of stride-128, keeps 8B align

__device__ __forceinline__ float lrelu(float v) {
    return v > 0.0f ? v : 0.2f * v;
}

// ---------------------------------------------------------------------------
// Kernel 1: h[b] = x[b] @ W  via V_WMMA_F32_16X16X4_F32 (exact f32 path).
// grid (NN/16, BB), block 256 (8 wave32s). Wave w -> 16x16 tile, cols [16w,16w+16).
// A frag (16x4, ISA layout): lanes 0-15 hold K=k0+0/1, lanes 16-31 hold K=k0+2/3.
// B frag (4x16): same K split across lane halves, N = lane&15.
// C/D (16x16): VGPR r -> row r (lanes 0-15) / row r+8 (lanes 16-31), col lane&15.
// ---------------------------------------------------------------------------
__global__ __launch_bounds__(256) void gemm_wmma_f32(const float* __restrict__ x,
                                                     const float* __restrict__ W,
                                                     float* __restrict__ h) {
    __shared__ float lx[16 * LPITCH];
    const int b   = blockIdx.y;
    const int m0  = blockIdx.x * 16;
    const int tid = threadIdx.x;

    // Stage 16x128 x-strip into LDS (coalesced B128-class loads).
    const float* xrow = x + ((size_t)b * NN + m0) * FF;
    for (int i = tid; i < 16 * FF; i += 256) {
        int r = i >> 7, kk = i & 127;
        lx[r * LPITCH + kk] = xrow[i];
    }
    __syncthreads();

    const int wave = tid >> 5;
    const int lane = tid & 31;
    const int hl   = lane >> 4;     // lane half
    const int l15  = lane & 15;
    const int n0   = wave * 16;

    v8f c = {};
    for (int k0 = 0; k0 < FF; k0 += 4) {
        const int ka = k0 + 2 * hl;
        v2f a, bf;
        a.x  = lx[l15 * LPITCH + ka];
        a.y  = lx[l15 * LPITCH + ka + 1];
        bf.x = W[(size_t)ka       * FF + n0 + l15];
        bf.y = W[(size_t)(ka + 1) * FF + n0 + l15];
        c = __builtin_amdgcn_wmma_f32_16x16x4_f32(
                /*neg_a=*/false, a, /*neg_b=*/false, bf,
                /*c_mod=*/(short)0, c, /*reuse_a=*/false, /*reuse_b=*/false);
    }

    float* hb = h + (size_t)b * NN * FF;
#pragma unroll
    for (int r = 0; r < 8; ++r) {
        const int row = m0 + r + (hl ? 8 : 0);
        hb[(size_t)row * FF + n0 + l15] = c[r];
    }
}

// ---------------------------------------------------------------------------
// Kernel 2: s[row] = h[row,:] . w_mlp + bias.  Wave-per-row, float4 + shfl reduce.
// grid = BB*NN/8, block 256 (8 waves).
// ---------------------------------------------------------------------------
__global__ __launch_bounds__(256) void score_kernel(const float* __restrict__ h,
                                                    const float* __restrict__ w_mlp,
                                                    const float* __restrict__ b_mlp,
                                                    float* __restrict__ s) {
    const int wave = threadIdx.x >> 5;
    const int lane = threadIdx.x & 31;
    const size_t row = (size_t)blockIdx.x * 8 + wave;
    const float4 hv = ((const float4*)(h + row * FF))[lane];
    const float4 wv = ((const float4*)w_mlp)[lane];
    float d = hv.x * wv.x + hv.y * wv.y + hv.z * wv.z + hv.w * wv.w;
#pragma unroll
    for (int off = 16; off > 0; off >>= 1) d += __shfl_xor(d, off, 32);
    if (lane == 0) s[row] = d + b_mlp[0];
}

// ---------------------------------------------------------------------------
// Kernel 3: smax[b] = max_j s[b,j].  grid = BB.
// ---------------------------------------------------------------------------
__global__ __launch_bounds__(256) void smax_kernel(const float* __restrict__ s,
                                                   float* __restrict__ smax) {
    __shared__ float red[256];
    const int b = blockIdx.x;
    float m = -INFINITY;
    for (int i = threadIdx.x; i < NN; i += 256) m = fmaxf(m, s[b * NN + i]);
    red[threadIdx.x] = m;
    __syncthreads();
    for (int st = 128; st > 0; st >>= 1) {
        if (threadIdx.x < st) red[threadIdx.x] = fmaxf(red[threadIdx.x], red[threadIdx.x + st]);
        __syncthreads();
    }
    if (threadIdx.x == 0) smax[b] = red[0];
}

// ---------------------------------------------------------------------------
// Kernel 4: Z[b,i] = sum_j exp(lrelu(s_i+s_j) - m_i), m_i = lrelu(s_i + smax_b).
// grid (NN/256, BB). s cached in LDS; inner loop is LDS-broadcast reads.
// ---------------------------------------------------------------------------
__global__ __launch_bounds__(256) void zrow_kernel(const float* __restrict__ s,
                                                   const float* __restrict__ smax,
                                                   float* __restrict__ Z) {
    __shared__ float ss[NN];
    const int b = blockIdx.y;
    for (int i = threadIdx.x; i < NN; i += 256) ss[i] = s[b * NN + i];
    __syncthreads();
    const int i = blockIdx.x * 256 + threadIdx.x;
    const float si = ss[i];
    const float m  = lrelu(si + smax[b]);
    float acc = 0.0f;
#pragma unroll 4
    for (int j = 0; j < NN; ++j) acc += __expf(lrelu(si + ss[j]) - m);
    Z[b * NN + i] = acc;
}

// ---------------------------------------------------------------------------
// Kernel 5: col[b,j] = sum_i exp(lrelu(s_i+s_j) - m_i) / Z_i
//           out[b,j,f] = lrelu(h[b,j,f] * col[b,j])
// grid (NN/256, BB). m_i and 1/Z_i precomputed into LDS; output written with a
// remapped index so stores stay fully coalesced.
// ---------------------------------------------------------------------------
__global__ __launch_bounds__(256) void colout_kernel(const float* __restrict__ s,
                                                     const float* __restrict__ smax,
                                                     const float* __restrict__ Z,
                                                     const float* __restrict__ h,
                                                     float* __restrict__ out) {
    __shared__ float ss[NN];
    __shared__ float mm[NN];
    __shared__ float iz[NN];
    __shared__ float colsh[256];
    const int b = blockIdx.y;
    const float smx = smax[b];
    for (int i = threadIdx.x; i < NN; i += 256) {
        const float si = s[b * NN + i];
        ss[i] = si;
        mm[i] = lrelu(si + smx);
        iz[i] = 1.0f / Z[b * NN + i];
    }
    __syncthreads();

    const int j  = blockIdx.x * 256 + threadIdx.x;
    const float sj = ss[j];
    float acc = 0.0f;
#pragma unroll 4
    for (int i = 0; i < NN; ++i)
        acc += __expf(lrelu(ss[i] + sj) - mm[i]) * iz[i];
    colsh[threadIdx.x] = acc;
    __syncthreads();

    const size_t base = ((size_t)b * NN + (size_t)blockIdx.x * 256) * FF;
    for (int idx = threadIdx.x; idx < 256 * FF; idx += 256) {
        const int r = idx >> 7;                 // local row
        out[base + idx] = lrelu(h[base + idx] * colsh[r]);
    }
}

// ---------------------------------------------------------------------------
extern "C" void kernel_launch(void* const* d_in, const int* in_sizes, int n_in,
                              void* d_out, int out_size, void* d_ws, size_t ws_size,
                              hipStream_t stream) {
    const float* x     = (const float*)d_in[0];   // [B,1,N,F_IN]
    const float* W     = (const float*)d_in[1];   // [F_IN,F_OUT]
    const float* w_mlp = (const float*)d_in[2];   // [F_OUT]
    const float* b_mlp = (const float*)d_in[3];   // [1]
    float* outp = (float*)d_out;                  // [B,1,N,F_OUT]

    float* ws   = (float*)d_ws;
    float* h    = ws;                             // B*N*F = 2,097,152 floats (8 MB)
    float* s    = h + (size_t)BB * NN * FF;       // B*N
    float* Z    = s + (size_t)BB * NN;            // B*N
    float* smax = Z + (size_t)BB * NN;            // B

    gemm_wmma_f32<<<dim3(NN / 16, BB), 256, 0, stream>>>(x, W, h);
    score_kernel <<<BB * NN / 8,       256, 0, stream>>>(h, w_mlp, b_mlp, s);
    smax_kernel  <<<BB,                256, 0, stream>>>(s, smax);
    zrow_kernel  <<<dim3(NN / 256, BB),256, 0, stream>>>(s, smax, Z);
    colout_kernel<<<dim3(NN / 256, BB),256, 0, stream>>>(s, smax, Z, h, outp);
}